// ReformerEncoder_22273700397103
// MI455X (gfx1250) — compile-verified
//
#include <hip/hip_runtime.h>
#include <hip/hip_bf16.h>

typedef float v2f __attribute__((ext_vector_type(2)));
typedef float v8f __attribute__((ext_vector_type(8)));

#define ROWS   131072      // B*S
#define SEQ    4096
#define BATCH  32
#define DIMM   64
#define NHASH  8
#define TOTAL  32768       // NHASH*SEQ
#define RSQRT_DH 0.3535533905932738f   // 8^-0.5

// ---- WMMA f32 16x16x4 wrapper (neg_a, A, neg_b, B, c_mod, C, reuse_a, reuse_b)
__device__ __forceinline__ v8f wmma4(v2f a, v2f b, v8f c) {
  return __builtin_amdgcn_wmma_f32_16x16x4_f32(false, a, false, b, (short)0, c, false, false);
}

// B fragment: 4x16 f32. VGPR0 = rows {k0, k0+2} (lanes 0-15 / 16-31), VGPR1 = rows {k0+1, k0+3}.
__device__ __forceinline__ v2f ld_b(const float* B, int ldb, int k0, int col, int hi) {
  int k = k0 + 2 * hi;
  v2f r = { B[k * ldb + col], B[(k + 1) * ldb + col] };
  return r;
}

// ---------------- embedding + fixed positional embedding ----------------
__global__ void embed_kernel(const int* __restrict__ xin, const float* __restrict__ emb,
                             float* __restrict__ x1, float* __restrict__ x2) {
  int idx = blockIdx.x * 256 + threadIdx.x;      // ROWS*DIMM threads
  int r = idx >> 6, d = idx & 63;
  int s = r & (SEQ - 1);
  int tok = xin[r];
  int i = d & 31;
  float freq = __expf(-(float)i * (9.210340371976184f / 32.f)); // 10000^(-i/32)
  float ang = (float)s * freq;
  float pos = (d < 32) ? sinf(ang) : cosf(ang);
  float val = emb[tok * DIMM + d] + pos;
  x1[idx] = val;
  x2[idx] = val;
}

// ---------------- layer norm: one wave32 per 64-wide row ----------------
__global__ void ln_kernel(const float* __restrict__ in, const float* __restrict__ g,
                          const float* __restrict__ bta, float* __restrict__ out) {
  int gtid = blockIdx.x * blockDim.x + threadIdx.x;
  int row  = gtid >> 5;
  int lane = threadIdx.x & 31;
  const float* rp = in + (size_t)row * DIMM;
  float a0 = rp[lane], a1 = rp[lane + 32];
  float s = a0 + a1;
  for (int m = 16; m; m >>= 1) s += __shfl_xor(s, m, 32);
  float mean = s * (1.f / 64.f);
  float d0 = a0 - mean, d1 = a1 - mean;
  float vs = d0 * d0 + d1 * d1;
  for (int m = 16; m; m >>= 1) vs += __shfl_xor(vs, m, 32);
  float rstd = 1.f / sqrtf(vs * (1.f / 64.f) + 1e-5f);
  float* op = out + (size_t)row * DIMM;
  op[lane]      = d0 * rstd * g[lane]      + bta[lane];
  op[lane + 32] = d1 * rstd * g[lane + 32] + bta[lane + 32];
}

// ---------------- QK and V projections: y(ROWSx64) @ W(64x64), WMMA ----------------
__global__ void qkv_kernel(const float* __restrict__ y, const float* __restrict__ wqk,
                           const float* __restrict__ wv, float* __restrict__ qkb,
                           float* __restrict__ vb) {
  int gtid = blockIdx.x * blockDim.x + threadIdx.x;
  int wid = gtid >> 5;                 // 16-row tile id
  int lane = threadIdx.x & 31;
  int r0 = wid << 4;
  int m = lane & 15, hi = lane >> 4;
  const float* ap = y + (size_t)(r0 + m) * DIMM + 2 * hi;
  v2f a[16];
#pragma unroll
  for (int kk = 0; kk < 16; ++kk) a[kk] = *(const v2f*)(ap + kk * 4);
  const float* Ws[2] = { wqk, wv };
  float* Os[2] = { qkb, vb };
  for (int o = 0; o < 2; ++o) {
    const float* W = Ws[o];
    float* O = Os[o];
#pragma unroll
    for (int nt = 0; nt < 4; ++nt) {
      int col = nt * 16 + m;
      v2f bf[16];
#pragma unroll
      for (int kk = 0; kk < 16; ++kk) bf[kk] = ld_b(W, DIMM, kk * 4, col, hi);
      v8f acc = {0.f, 0.f, 0.f, 0.f, 0.f, 0.f, 0.f, 0.f};
#pragma unroll
      for (int kk = 0; kk < 16; ++kk) acc = wmma4(a[kk], bf[kk], acc);
#pragma unroll
      for (int j = 0; j < 8; ++j)
        O[(size_t)(r0 + j + 8 * hi) * DIMM + col] = acc[j];
    }
  }
}

// ---------------- LSH bucketing: fused rotation GEMM (K=8) + argmax over 4096 ----------------
// Wave handles 32 query rows (two A tiles sharing each B fragment); sweeps 128 N-tiles of the
// 2048 rotations with WMMA, software-pipelining B loads. argmax over concat([r,-r]) is done as
// (|r|, index = c + (r<0)*2048) with min-index tie-break (exactly first-max semantics).
__global__ void bucket_kernel(const float* __restrict__ qk, const float* __restrict__ rot,
                              unsigned* __restrict__ keys) {
  int gtid = blockIdx.x * blockDim.x + threadIdx.x;
  int wid = gtid >> 5;                  // [0, 256*8*128)
  int lane = threadIdx.x & 31;
  int stile = wid & 127;
  int n = (wid >> 7) & 7;
  int bh = wid >> 10;
  int b = bh >> 3, h = bh & 7;
  int s0 = stile * 32;
  int m = lane & 15, hi = lane >> 4;
  const float* qp0 = qk + (size_t)(b * SEQ + s0 + m) * DIMM + h * 8 + 2 * hi;
  const float* qp1 = qp0 + (size_t)16 * DIMM;
  v2f a0 = *(const v2f*)(qp0);
  v2f a1 = *(const v2f*)(qp0 + 4);
  v2f a2 = *(const v2f*)(qp1);
  v2f a3 = *(const v2f*)(qp1 + 4);
  const float* rp = rot + (size_t)n * 8 * 2048;
  const float* r0p = rp + (2 * hi) * 2048 + m;        // rows {2hi, 2hi+1}
  const float* r1p = rp + (4 + 2 * hi) * 2048 + m;    // rows {4+2hi, 5+2hi}
  float bestv[16];
  int   besti[16];
#pragma unroll
  for (int j = 0; j < 16; ++j) { bestv[j] = -3.0e38f; besti[j] = 0; }
  v2f b0 = { r0p[0], r0p[2048] };
  v2f b1 = { r1p[0], r1p[2048] };
  for (int nt = 0; nt < 128; ++nt) {
    int c = nt * 16 + m;
    v2f nb0, nb1;
    if (nt < 127) {
      nb0 = v2f{ r0p[(nt + 1) * 16], r0p[(nt + 1) * 16 + 2048] };
      nb1 = v2f{ r1p[(nt + 1) * 16], r1p[(nt + 1) * 16 + 2048] };
    }
    v8f acc0 = {0.f, 0.f, 0.f, 0.f, 0.f, 0.f, 0.f, 0.f};
    acc0 = wmma4(a0, b0, acc0);
    acc0 = wmma4(a1, b1, acc0);
    v8f acc1 = {0.f, 0.f, 0.f, 0.f, 0.f, 0.f, 0.f, 0.f};
    acc1 = wmma4(a2, b0, acc1);
    acc1 = wmma4(a3, b1, acc1);
#pragma unroll
    for (int j = 0; j < 8; ++j) {
      float vv = acc0[j];
      float av = fabsf(vv);
      int ni = c + ((vv < 0.f) ? 2048 : 0);
      if (av > bestv[j] || (av == bestv[j] && ni < besti[j])) { bestv[j] = av; besti[j] = ni; }
      float vw = acc1[j];
      float aw = fabsf(vw);
      int nj = c + ((vw < 0.f) ? 2048 : 0);
      if (aw > bestv[8 + j] || (aw == bestv[8 + j] && nj < besti[8 + j])) {
        bestv[8 + j] = aw; besti[8 + j] = nj;
      }
    }
    b0 = nb0; b1 = nb1;
  }
  // reduce across the 16 lanes sharing each row group
#pragma unroll
  for (int msk = 1; msk < 16; msk <<= 1) {
#pragma unroll
    for (int j = 0; j < 16; ++j) {
      float ov = __shfl_xor(bestv[j], msk, 32);
      int   oi = __shfl_xor(besti[j], msk, 32);
      if (ov > bestv[j] || (ov == bestv[j] && oi < besti[j])) { bestv[j] = ov; besti[j] = oi; }
    }
  }
  if (m == 0) {
    unsigned* kp = keys + ((size_t)bh * NHASH + n) * SEQ;
#pragma unroll
    for (int j = 0; j < 8; ++j) {
      int p0 = s0 + j + 8 * hi;
      kp[p0] = ((unsigned)besti[j] << 12) | (unsigned)p0;
      int p1 = s0 + 16 + j + 8 * hi;
      kp[p1] = ((unsigned)besti[8 + j] << 12) | (unsigned)p1;
    }
  }
}

// ---------------- per-(bh,hash) bitonic sort of 4096 unique keys in LDS ----------------
__global__ void sort_kernel(unsigned* __restrict__ keys) {
  __shared__ unsigned sk[SEQ];
  size_t base = (size_t)blockIdx.x * SEQ;
  for (int i = threadIdx.x; i < SEQ; i += 256) sk[i] = keys[base + i];
  __syncthreads();
  for (int k = 2; k <= SEQ; k <<= 1) {
    for (int j = k >> 1; j > 0; j >>= 1) {
      for (int i = threadIdx.x; i < SEQ; i += 256) {
        int ixj = i ^ j;
        if (ixj > i) {
          bool up = ((i & k) == 0);
          unsigned a = sk[i], c = sk[ixj];
          if ((a > c) == up) { sk[i] = c; sk[ixj] = a; }
        }
      }
      __syncthreads();
    }
  }
  for (int i = threadIdx.x; i < SEQ; i += 256) keys[base + i] = sk[i] & 4095u; // store st
}

// ---------------- pass A: per sorted position, compute logsumexp + neighbor ----------------
__global__ void passA_kernel(const float* __restrict__ qk, const unsigned* __restrict__ st,
                             const int* __restrict__ xin, float* __restrict__ lg,
                             int* __restrict__ nbb) {
  int gtid = blockIdx.x * blockDim.x + threadIdx.x;
  int bh = gtid >> 15;
  int i  = gtid & (TOTAL - 1);
  int b = bh >> 3, h = bh & 7;
  const unsigned* stp = st + (size_t)bh * TOTAL;
  int p  = (int)stp[i];
  int p2 = (int)stp[(i + TOTAL - 1) & (TOTAL - 1)];
  const float* q  = qk + (size_t)(b * SEQ + p)  * DIMM + h * 8;
  const float* k2 = qk + (size_t)(b * SEQ + p2) * DIMM + h * 8;
  float nrm2 = 0.f, dt = 0.f;
#pragma unroll
  for (int c = 0; c < 8; ++c) { float kv = k2[c]; nrm2 += kv * kv; dt += q[c] * kv; }
  float d1;
  if (p == p2) d1 = -50000.f;
  else if (xin[b * SEQ + p] != 0 && xin[b * SEQ + p2] != 0)
    d1 = (dt / fmaxf(sqrtf(nrm2), 1e-12f)) * RSQRT_DH;
  else d1 = -1e9f;
  const float d0 = -50000.f;
  float mx = fmaxf(d0, d1);
  float lse = mx + logf(expf(d0 - mx) + expf(d1 - mx));
  int n = i >> 12;
  lg [(size_t)bh * TOTAL + n * SEQ + p] = lse;
  nbb[(size_t)bh * TOTAL + n * SEQ + p] = p2;
}

// ---------------- combine: softmax over hash rounds, recompute 2-key attention ----------------
__global__ void combine_kernel(const float* __restrict__ qk, const float* __restrict__ v,
                               const int* __restrict__ xin, const float* __restrict__ lg,
                               const int* __restrict__ nbb, float* __restrict__ ob) {
  int gtid = blockIdx.x * blockDim.x + threadIdx.x;
  int bh = gtid >> 12;
  int p  = gtid & (SEQ - 1);
  int b = bh >> 3, h = bh & 7;
  float lses[NHASH], pe[NHASH];
  float mx = -3.0e38f;
#pragma unroll
  for (int n = 0; n < NHASH; ++n) {
    lses[n] = lg[(size_t)bh * TOTAL + n * SEQ + p];
    mx = fmaxf(mx, lses[n]);
  }
  float se = 0.f;
#pragma unroll
  for (int n = 0; n < NHASH; ++n) { pe[n] = expf(lses[n] - mx); se += pe[n]; }
  float inv = 1.f / se;
  const float* q  = qk + (size_t)(b * SEQ + p) * DIMM + h * 8;
  const float* vq = v  + (size_t)(b * SEQ + p) * DIMM + h * 8;
  bool mq = xin[b * SEQ + p] != 0;
  float acc[8] = {0.f, 0.f, 0.f, 0.f, 0.f, 0.f, 0.f, 0.f};
#pragma unroll
  for (int n = 0; n < NHASH; ++n) {
    int p2 = nbb[(size_t)bh * TOTAL + n * SEQ + p];
    const float* k2 = qk + (size_t)(b * SEQ + p2) * DIMM + h * 8;
    const float* v2 = v  + (size_t)(b * SEQ + p2) * DIMM + h * 8;
    float nrm2 = 0.f, dt = 0.f;
#pragma unroll
    for (int c = 0; c < 8; ++c) { float kv = k2[c]; nrm2 += kv * kv; dt += q[c] * kv; }
    float d1;
    if (p == p2) d1 = -50000.f;
    else if (mq && xin[b * SEQ + p2] != 0) d1 = (dt / fmaxf(sqrtf(nrm2), 1e-12f)) * RSQRT_DH;
    else d1 = -1e9f;
    float e0 = expf(-50000.f - lses[n]);
    float e1 = expf(d1 - lses[n]);
    float w = pe[n] * inv;
#pragma unroll
    for (int c = 0; c < 8; ++c) acc[c] += w * (e0 * vq[c] + e1 * v2[c]);
  }
  float* op = ob + (size_t)(b * SEQ + p) * DIMM + h * 8;
#pragma unroll
  for (int c = 0; c < 8; ++c) op[c] = acc[c];
}

// ---------------- output projection + residual: x1 += o @ wo + b_o ----------------
__global__ void oproj_kernel(const float* __restrict__ o, const float* __restrict__ wo,
                             const float* __restrict__ bo, float* __restrict__ x1) {
  int gtid = blockIdx.x * blockDim.x + threadIdx.x;
  int wid = gtid >> 5;
  int lane = threadIdx.x & 31;
  int r0 = wid << 4;
  int m = lane & 15, hi = lane >> 4;
  const float* ap = o + (size_t)(r0 + m) * DIMM + 2 * hi;
  v2f a[16];
#pragma unroll
  for (int kk = 0; kk < 16; ++kk) a[kk] = *(const v2f*)(ap + kk * 4);
#pragma unroll
  for (int nt = 0; nt < 4; ++nt) {
    int col = nt * 16 + m;
    v2f bf[16];
#pragma unroll
    for (int kk = 0; kk < 16; ++kk) bf[kk] = ld_b(wo, DIMM, kk * 4, col, hi);
    v8f acc = {0.f, 0.f, 0.f, 0.f, 0.f, 0.f, 0.f, 0.f};
#pragma unroll
    for (int kk = 0; kk < 16; ++kk) acc = wmma4(a[kk], bf[kk], acc);
    float bb = bo[col];
#pragma unroll
    for (int j = 0; j < 8; ++j) {
      size_t idx = (size_t)(r0 + j + 8 * hi) * DIMM + col;
      x1[idx] = x1[idx] + acc[j] + bb;
    }
  }
}

// ---------------- fused FFN: x2 += gelu(y@W1+b1)@W2+b2 (intermediate in LDS only) ----------------
#define FFN_WAVES 2
__global__ void ffn_kernel(const float* __restrict__ y, const float* __restrict__ w1,
                           const float* __restrict__ b1, const float* __restrict__ w2,
                           const float* __restrict__ b2, float* __restrict__ x2) {
  __shared__ float tl[FFN_WAVES][16][258];
  int wv = threadIdx.x >> 5, lane = threadIdx.x & 31;
  int r0 = (blockIdx.x * FFN_WAVES + wv) << 4;
  int m = lane & 15, hi = lane >> 4;
  const float* ap = y + (size_t)(r0 + m) * DIMM + 2 * hi;
  v2f a[16];
#pragma unroll
  for (int kk = 0; kk < 16; ++kk) a[kk] = *(const v2f*)(ap + kk * 4);
#pragma unroll
  for (int nt = 0; nt < 16; ++nt) {
    int col = nt * 16 + m;
    v2f bf[16];
#pragma unroll
    for (int kk = 0; kk < 16; ++kk) bf[kk] = ld_b(w1, 256, kk * 4, col, hi);
    v8f acc = {0.f, 0.f, 0.f, 0.f, 0.f, 0.f, 0.f, 0.f};
#pragma unroll
    for (int kk = 0; kk < 16; ++kk) acc = wmma4(a[kk], bf[kk], acc);
    float bb = b1[col];
#pragma unroll
    for (int j = 0; j < 8; ++j) {
      float vv = acc[j] + bb;
      tl[wv][j + 8 * hi][col] = 0.5f * vv * (1.f + erff(vv * 0.7071067811865476f));
    }
  }
  __syncthreads();
#pragma unroll
  for (int nt = 0; nt < 4; ++nt) {
    int col = nt * 16 + m;
    v8f acc = {0.f, 0.f, 0.f, 0.f, 0.f, 0.f, 0.f, 0.f};
#pragma unroll
    for (int ck = 0; ck < 4; ++ck) {           // K=256 in 4 chunks of 64
      v2f bf[16];
#pragma unroll
      for (int kk = 0; kk < 16; ++kk)
        bf[kk] = ld_b(w2, DIMM, (ck * 16 + kk) * 4, col, hi);
#pragma unroll
      for (int kk = 0; kk < 16; ++kk) {
        int k = (ck * 16 + kk) * 4 + 2 * hi;
        v2f af = *(const v2f*)&tl[wv][m][k];
        acc = wmma4(af, bf[kk], acc);
      }
    }
    float bb = b2[col];
#pragma unroll
    for (int j = 0; j < 8; ++j) {
      size_t idx = (size_t)(r0 + j + 8 * hi) * DIMM + col;
      x2[idx] = x2[idx] + acc[j] + bb;
    }
  }
}

// ---------------- classifier head: rep = 0.5*(x1+x2)[:,0,:], relu MLP ----------------
__global__ void cls_kernel(const float* __restrict__ x1, const float* __restrict__ x2,
                           const float* __restrict__ w1, const float* __restrict__ b1,
                           const float* __restrict__ w2, const float* __restrict__ b2,
                           float* __restrict__ out) {
  __shared__ float rep[64];
  __shared__ float hid[32];
  int b = blockIdx.x, t = threadIdx.x;              // 64 threads
  size_t row = (size_t)b * SEQ * DIMM;
  float rv = 0.5f * (x1[row + t] + x2[row + t]);
  rep[t] = rv;
  out[64 + b * 64 + t] = rv;                        // rep part of output tuple
  __syncthreads();
  if (t < 32) {
    float s = b1[t];
    for (int d = 0; d < 64; ++d) s += rep[d] * w1[d * 32 + t];
    hid[t] = fmaxf(s, 0.f);
  }
  __syncthreads();
  if (t < 2) {
    float s = b2[t];
    for (int j = 0; j < 32; ++j) s += hid[j] * w2[j * 2 + t];
    out[b * 2 + t] = s;                             // logits part of output tuple
  }
}

extern "C" void kernel_launch(void* const* d_in, const int* in_sizes, int n_in,
                              void* d_out, int out_size, void* d_ws, size_t ws_size,
                              hipStream_t stream) {
  (void)in_sizes; (void)n_in; (void)out_size; (void)ws_size;
  const int*   xin  = (const int*)  d_in[0];
  const float* emb  = (const float*)d_in[1];
  const float* rot  = (const float*)d_in[2];
  const float* w_qk = (const float*)d_in[3];
  const float* w_v  = (const float*)d_in[4];
  const float* w_o  = (const float*)d_in[5];
  const float* b_o  = (const float*)d_in[6];
  const float* ln1g = (const float*)d_in[7];
  const float* ln1b = (const float*)d_in[8];
  const float* ffw1 = (const float*)d_in[9];
  const float* ffb1 = (const float*)d_in[10];
  const float* ffw2 = (const float*)d_in[11];
  const float* ffb2 = (const float*)d_in[12];
  const float* ln2g = (const float*)d_in[13];
  const float* ln2b = (const float*)d_in[14];
  const float* cw1  = (const float*)d_in[15];
  const float* cb1  = (const float*)d_in[16];
  const float* cw2  = (const float*)d_in[17];
  const float* cb2  = (const float*)d_in[18];
  float* out = (float*)d_out;

  const size_t RD = (size_t)ROWS * DIMM;            // 8388608 elements per buffer
  float* ws = (float*)d_ws;
  float*    x1   = ws;
  float*    x2   = ws + RD;
  float*    yb   = ws + 2 * RD;
  float*    qkb  = ws + 3 * RD;
  float*    vb   = ws + 4 * RD;
  float*    ob   = ws + 5 * RD;
  unsigned* keys = (unsigned*)(ws + 6 * RD);
  int*      nbb  = (int*)(ws + 7 * RD);
  float*    lg   = ws + 8 * RD;

  embed_kernel<<<RD / 256, 256, 0, stream>>>(xin, emb, x1, x2);
  for (int l = 0; l < 2; ++l) {
    ln_kernel<<<(ROWS * 32) / 256, 256, 0, stream>>>(x2, ln1g + l * 64, ln1b + l * 64, yb);
    qkv_kernel<<<((ROWS / 16) * 32) / 256, 256, 0, stream>>>(
        yb, w_qk + l * 64 * 64, w_v + l * 64 * 64, qkb, vb);
    bucket_kernel<<<(256 * 8 * 128 * 32) / 256, 256, 0, stream>>>(
        qkb, rot + (size_t)l * NHASH * 8 * 2048, keys);
    sort_kernel<<<256 * NHASH, 256, 0, stream>>>(keys);
    passA_kernel<<<(256 * TOTAL) / 256, 256, 0, stream>>>(qkb, keys, xin, lg, nbb);
    combine_kernel<<<(256 * SEQ) / 256, 256, 0, stream>>>(qkb, vb, xin, lg, nbb, ob);
    oproj_kernel<<<((ROWS / 16) * 32) / 256, 256, 0, stream>>>(
        ob, w_o + l * 64 * 64, b_o + l * 64, x1);
    ln_kernel<<<(ROWS * 32) / 256, 256, 0, stream>>>(x1, ln2g + l * 64, ln2b + l * 64, yb);
    ffn_kernel<<<ROWS / (16 * FFN_WAVES), 32 * FFN_WAVES, 0, stream>>>(
        yb, ffw1 + l * 64 * 256, ffb1 + l * 256, ffw2 + l * 256 * 64, ffb2 + l * 64, x2);
  }
  cls_kernel<<<BATCH, 64, 0, stream>>>(x1, x2, cw1, cb1, cw2, cb2, out);
}